// ShortConv_89807766159679
// MI455X (gfx1250) — compile-verified
//
#include <hip/hip_runtime.h>

// Problem constants (match reference).
#define DIM   2048
#define KSZ   4
#define BATCH 4
#define SEQ   4096

#define TPB    256                 // threads per block (8 waves on wave32)
#define CHUNK  128                 // sequence rows per block-chunk
#define NCHUNK (SEQ / CHUNK)       // 32
#define PFDIST 16                  // prefetch distance (rows ahead)

typedef float v4f __attribute__((ext_vector_type(4)));
// Exact pointee type of the async-copy builtin params (per hipcc diagnostic):
// "__attribute__((__vector_size__(4 * sizeof(int)))) int"
typedef int v4i __attribute__((__vector_size__(4 * sizeof(int))));
typedef __attribute__((address_space(1))) v4i gv4i;   // global (prints as __device__)
typedef __attribute__((address_space(3))) v4i lv4i;   // LDS    (prints as __shared__)

// gfx1250 async global->LDS copy path (guarded: falls back to plain loads).
#if defined(__has_builtin)
#if __has_builtin(__builtin_amdgcn_global_load_async_to_lds_b128)
#define USE_ASYNC_LDS 1
#endif
#endif
#ifndef USE_ASYNC_LDS
#define USE_ASYNC_LDS 0
#endif

__global__ __launch_bounds__(TPB) void ShortConv_89807766159679_kernel(
    const float* __restrict__ x,     // (B, S, D)
    const float* __restrict__ w,     // (D, 1, K) -> row d = 4 contiguous floats
    const float* __restrict__ bias,  // (D,)
    float* __restrict__ y)           // (B, S, D)
{
    const int tid = threadIdx.x;
    const int g   = blockIdx.x * TPB + tid;        // channel group [0, DIM/4)
    const int d   = g * 4;                         // first channel of group
    const int bc  = blockIdx.y;                    // b * NCHUNK + chunk
    const int b   = bc / NCHUNK;
    const int c   = bc % NCHUNK;
    const int s0  = c * CHUNK;

    // ---- Stage per-thread weights (4 rows x 4 taps = 64B) + bias (16B) ----
    v4f w0, w1, w2, w3, bv;
#if USE_ASYNC_LDS
    __shared__ __align__(16) float lw[TPB * 16];   // 16 KB
    __shared__ __align__(16) float lb[TPB * 4];    //  4 KB
    {
        const float* gw = w + (size_t)d * KSZ;     // 16 contiguous floats
        __builtin_amdgcn_global_load_async_to_lds_b128(
            (gv4i*)(gw + 0),  (lv4i*)&lw[tid * 16 + 0], 0, 0);
        __builtin_amdgcn_global_load_async_to_lds_b128(
            (gv4i*)(gw + 4),  (lv4i*)&lw[tid * 16 + 4], 0, 0);
        __builtin_amdgcn_global_load_async_to_lds_b128(
            (gv4i*)(gw + 8),  (lv4i*)&lw[tid * 16 + 8], 0, 0);
        __builtin_amdgcn_global_load_async_to_lds_b128(
            (gv4i*)(gw + 12), (lv4i*)&lw[tid * 16 + 12], 0, 0);
        __builtin_amdgcn_global_load_async_to_lds_b128(
            (gv4i*)(bias + d), (lv4i*)&lb[tid * 4], 0, 0);
#if __has_builtin(__builtin_amdgcn_s_wait_asynccnt)
        __builtin_amdgcn_s_wait_asynccnt(0);
#else
        asm volatile("s_wait_asynccnt 0" ::: "memory");
#endif
        w0 = *(const v4f*)&lw[tid * 16 + 0];
        w1 = *(const v4f*)&lw[tid * 16 + 4];
        w2 = *(const v4f*)&lw[tid * 16 + 8];
        w3 = *(const v4f*)&lw[tid * 16 + 12];
        bv = *(const v4f*)&lb[tid * 4];
    }
#else
    {
        const v4f* wv = (const v4f*)w;             // wv[i] = taps of channel i
        w0 = wv[d + 0];
        w1 = wv[d + 1];
        w2 = wv[d + 2];
        w3 = wv[d + 3];
        bv = ((const v4f*)bias)[g];
    }
#endif

    // Transpose 4x4 weight block: t_k[j] = tap k of channel d+j, so the inner
    // loop is pure vector FMAs.
    const v4f t0 = (v4f){w0[0], w1[0], w2[0], w3[0]};  // multiplies x[s-3]
    const v4f t1 = (v4f){w0[1], w1[1], w2[1], w3[1]};  // multiplies x[s-2]
    const v4f t2 = (v4f){w0[2], w1[2], w2[2], w3[2]};  // multiplies x[s-1]
    const v4f t3 = (v4f){w0[3], w1[3], w2[3], w3[3]};  // multiplies x[s]

    // ---- Sliding-window history (causal left-pad with zeros) ----
    const size_t rowstride = (size_t)DIM;
    const float* xb = x + ((size_t)b * SEQ) * rowstride + d;
    float*       yb = y + ((size_t)b * SEQ) * rowstride + d;

    v4f p3 = (v4f)0.f, p2 = (v4f)0.f, p1 = (v4f)0.f;
    if (s0 > 0) {  // CHUNK >= 3, so the whole halo is in-bounds
        p3 = *(const v4f*)(xb + (size_t)(s0 - 3) * rowstride);
        p2 = *(const v4f*)(xb + (size_t)(s0 - 2) * rowstride);
        p1 = *(const v4f*)(xb + (size_t)(s0 - 1) * rowstride);
    }

    // ---- Stream CHUNK rows: 1 b128 load + 4 vec-FMA + 1 NT b128 store ----
#pragma unroll 4
    for (int s = s0; s < s0 + CHUNK; ++s) {
        const float* xp = xb + (size_t)s * rowstride;
        if (s + PFDIST < SEQ)
            __builtin_prefetch(xp + (size_t)PFDIST * rowstride, 0, 1);
        v4f cur = *(const v4f*)xp;
        v4f acc = bv + t0 * p3 + t1 * p2 + t2 * p1 + t3 * cur;
        __builtin_nontemporal_store(acc, (v4f*)(yb + (size_t)s * rowstride));
        p3 = p2;
        p2 = p1;
        p1 = cur;
    }
}

extern "C" void kernel_launch(void* const* d_in, const int* in_sizes, int n_in,
                              void* d_out, int out_size, void* d_ws, size_t ws_size,
                              hipStream_t stream) {
    (void)in_sizes; (void)n_in; (void)out_size; (void)d_ws; (void)ws_size;
    const float* x    = (const float*)d_in[0];  // (B, S, D) fp32
    const float* wgt  = (const float*)d_in[1];  // (D, 1, K) fp32
    const float* bias = (const float*)d_in[2];  // (D,) fp32
    float*       out  = (float*)d_out;          // (B, S, D) fp32

    dim3 grid((DIM / 4) / TPB, BATCH * NCHUNK);  // (2, 128)
    dim3 block(TPB);
    ShortConv_89807766159679_kernel<<<grid, block, 0, stream>>>(x, wgt, bias, out);
}